// SingleRenderer_73839077753381
// MI455X (gfx1250) — compile-verified
//
#include <hip/hip_runtime.h>
#include <hip/hip_bf16.h>
#include <stdint.h>

// NeRF single-pass volume renderer for MI455X (gfx1250, wave32).
// Thread-per-ray (32 rays per wave), samples streamed in chunks of 16 through
// a double-buffered LDS tile staged by GLOBAL_LOAD_ASYNC_TO_LDS_B128 in a
// transposed [group][ray] layout (conflict-free b128 LDS reads, private scan).

#define WAVES_PER_BLOCK 4
#define BLOCK_THREADS   128
#define GRID_BLOCKS     2048
#define CHUNK_SAMPLES   16
#define N_CHUNKS        8      // 128 samples / 16
#define BUF_FLOATS      2048   // per buffer: sigma 4*32*4 + radiance 12*32*4 floats = 8 KB
#define FAR_DIST        1e10f
#define EPS_T           1e-10f

// ---- CDNA5 async global->LDS copy (ASYNCcnt path) ----
__device__ __forceinline__ void async_copy_b128(uint32_t lds_off, const void* gaddr) {
  asm volatile("global_load_async_to_lds_b128 %0, %1, off"
               :: "v"(lds_off), "v"(gaddr)
               : "memory");
}
__device__ __forceinline__ void wait_async_le16() {
#if __has_builtin(__builtin_amdgcn_s_wait_asynccnt)
  __builtin_amdgcn_s_wait_asynccnt(16);
#else
  asm volatile("s_wait_asynccnt 0x10" ::: "memory");
#endif
}
__device__ __forceinline__ void wait_async_0() {
#if __has_builtin(__builtin_amdgcn_s_wait_asynccnt)
  __builtin_amdgcn_s_wait_asynccnt(0);
#else
  asm volatile("s_wait_asynccnt 0x0" ::: "memory");
#endif
}
__device__ __forceinline__ void wait_ds_0() {
  asm volatile("s_wait_dscnt 0x0" ::: "memory");
}

__global__ __launch_bounds__(BLOCK_THREADS)
void nerf_render_kernel(const float* __restrict__ near_,
                        const float* __restrict__ far_,
                        const float* __restrict__ sigma,
                        const float* __restrict__ radiance,
                        float* __restrict__ out,
                        int n_rays) {
  // 8 KB per buffer, double buffered, per wave -> 64 KB per 4-wave block.
  __shared__ __align__(16) float lds_buf[WAVES_PER_BLOCK][2][BUF_FLOATS];

  const int lane   = threadIdx.x & 31;
  const int wid    = threadIdx.x >> 5;
  const int gwave  = blockIdx.x * WAVES_PER_BLOCK + wid;
  const int total_waves = gridDim.x * WAVES_PER_BLOCK;
  const int nblocks = (n_rays + 31) >> 5;
  if (n_rays <= 0) return;

  uint32_t lbase[2];
  lbase[0] = (uint32_t)(uintptr_t)(&lds_buf[wid][0][0]);
  lbase[1] = (uint32_t)(uintptr_t)(&lds_buf[wid][1][0]);

  for (int rb = gwave; rb < nblocks; rb += total_waves) {
    const int ray_base = rb << 5;
    const int ray      = ray_base + lane;
    const int ray_c    = (ray < n_rays) ? ray : (n_rays - 1);  // clamp OOB lanes

    // Stage chunk c (16 samples) of this wave's 32 rays into buffer `buf`.
    // LDS layout (float4 units): sigma groups g=0..3 at (g*32+ray)*16B,
    // radiance groups j=0..11 at 2048 + (j*32+ray)*16B  (sample-major floats).
    auto issue = [&](int buf, int c) {
      const uint32_t lb = lbase[buf];
      const uint32_t lo = (uint32_t)lane * 16u;
      const char* gs = (const char*)(sigma    + (size_t)ray_c * 128 + (size_t)c * 16);
      const char* gr = (const char*)(radiance + (size_t)ray_c * 384 + (size_t)c * 48);
#pragma unroll
      for (int j = 0; j < 4; ++j)
        async_copy_b128(lb + (uint32_t)(j * 512) + lo, gs + j * 16);
#pragma unroll
      for (int j = 0; j < 12; ++j)
        async_copy_b128(lb + 2048u + (uint32_t)(j * 512) + lo, gr + j * 16);
    };

    const float nearv = near_[ray_c];
    const float farv  = far_[ray_c];

    float T = 1.0f;                       // running transmittance (private scan)
    float sr = 0.f, sg = 0.f, sb = 0.f, sd = 0.f, sa = 0.f;

    int cur = 0;
    issue(0, 0);

    for (int c = 0; c < N_CHUNKS; ++c) {
      if (c + 1 < N_CHUNKS) {
        wait_ds_0();                      // previous reads of buf cur^1 fully retired
        issue(cur ^ 1, c + 1);
        wait_async_le16();                // chunk c's 16 async ops complete (in-order)
      } else {
        wait_async_0();
      }

      const float* bufp = &lds_buf[wid][cur][0];
#pragma unroll
      for (int g = 0; g < 4; ++g) {
        const float4 sv = *(const float4*)(bufp + (size_t)(g * 32 + lane) * 4);
        const float4 ra = *(const float4*)(bufp + 512 + (size_t)((3 * g + 0) * 32 + lane) * 4);
        const float4 rc0 = *(const float4*)(bufp + 512 + (size_t)((3 * g + 1) * 32 + lane) * 4);
        const float4 rc1 = *(const float4*)(bufp + 512 + (size_t)((3 * g + 2) * 32 + lane) * 4);
        const int sbase = c * CHUNK_SAMPLES + g * 4;

        const float svk[4] = {sv.x, sv.y, sv.z, sv.w};
        const float rr[4]  = {ra.x, ra.w, rc0.z, rc1.y};
        const float rg[4]  = {ra.y, rc0.x, rc0.w, rc1.z};
        const float rbv[4] = {ra.z, rc0.y, rc1.x, rc1.w};

#pragma unroll
        for (int k = 0; k < 4; ++k) {
          const int s    = sbase + k;
          const float t  = (float)s * (1.0f / 127.0f);
          const float d  = nearv * (1.0f - t) + farv * t;          // d_final[s]
          float dist;
          if (s == 127) {
            dist = FAR_DIST;
          } else {
            const float t2 = (float)(s + 1) * (1.0f / 127.0f);
            dist = (nearv * (1.0f - t2) + farv * t2) - d;
          }
          const float e = __expf(-fmaxf(svk[k], 0.0f) * dist);
          const float a = 1.0f - e;                                 // alpha
          const float w = a * T;                                    // weight
          T *= (1.0f - a) + EPS_T;                                  // cumprod factor
          sr = fmaf(w, rr[k],  sr);
          sg = fmaf(w, rg[k],  sg);
          sb = fmaf(w, rbv[k], sb);
          sd = fmaf(w, d, sd);
          sa += w;
        }
      }
      cur ^= 1;
    }

    if (ray < n_rays) {
      out[3 * (size_t)ray + 0]        = sr;   // rgb_map
      out[3 * (size_t)ray + 1]        = sg;
      out[3 * (size_t)ray + 2]        = sb;
      out[(size_t)3 * n_rays + ray]   = sd;   // depth_map
      out[(size_t)4 * n_rays + ray]   = sa;   // acc_map
    }
  }
}

extern "C" void kernel_launch(void* const* d_in, const int* in_sizes, int n_in,
                              void* d_out, int out_size, void* d_ws, size_t ws_size,
                              hipStream_t stream) {
  const float* near_    = (const float*)d_in[0];
  const float* far_     = (const float*)d_in[1];
  const float* sigma    = (const float*)d_in[2];
  const float* radiance = (const float*)d_in[3];
  float* out = (float*)d_out;
  const int n_rays = in_sizes[0];

  (void)n_in; (void)out_size; (void)d_ws; (void)ws_size;

  const int nblocks    = (n_rays + 31) >> 5;
  const int wgs_needed = (nblocks + WAVES_PER_BLOCK - 1) / WAVES_PER_BLOCK;
  const int grid       = wgs_needed < GRID_BLOCKS ? (wgs_needed > 0 ? wgs_needed : 1)
                                                  : GRID_BLOCKS;

  nerf_render_kernel<<<dim3(grid), dim3(BLOCK_THREADS), 0, stream>>>(
      near_, far_, sigma, radiance, out, n_rays);
}